// ScatterAttention1d_23304492548279
// MI455X (gfx1250) — compile-verified
//
#include <hip/hip_runtime.h>
#include <math.h>

#define NUM_SAMPLES 16
#define BATCH 64
#define MAX_OFFSET_F 32.0f
#define CHUNK 1024      // floats per column chunk staged to LDS (4 KB)
#define ROWGRP 16       // batch rows processed per block (nw reuse factor)

typedef float vfloat4 __attribute__((ext_vector_type(4)));

// ws float layout:
// [0..15]   warped[k]
// [16..31]  raw_weight[k]
// [32..47]  Fk[k]     (int bits: floor(d + warped[k]))
// [48]      d (clipped deformation)
// [49]      Mloop     (int bits: boundary brute-force window)
// [51]      inv_sum
// [64 .. 64+L)              nw map (float)
// [64+L .. 64+L+numBlocks)  block partial sums

__device__ __forceinline__ float softplus_f(float x) {
    // stable: max(x,0) + log1p(exp(-|x|))  (matches jax.nn.softplus numerically)
    return fmaxf(x, 0.0f) + log1pf(expf(-fabsf(x)));
}

__global__ void prep_kernel(const float* __restrict__ bd, const float* __restrict__ bs,
                            const float* __restrict__ bbf, const float* __restrict__ bbb,
                            const float* __restrict__ bstr, const float* __restrict__ baf,
                            const float* __restrict__ bab, const float* __restrict__ bias,
                            float* __restrict__ hdr) {
    if (threadIdx.x != 0 || blockIdx.x != 0) return;
    float deformation = fminf(fmaxf(bd[0], -MAX_OFFSET_F), MAX_OFFSET_F);
    float stride   = softplus_f(bs[0]);
    float beta_f   = softplus_f(bbf[0]);
    float beta_b   = softplus_f(bbb[0]);
    float strength = softplus_f(bstr[0]);
    float alpha_f  = softplus_f(baf[0]);
    float alpha_b  = softplus_f(bab[0]);
    int* hdri = (int*)hdr;
    float cmax = 0.0f;
    for (int i = 0; i < NUM_SAMPLES; ++i) {
        float k  = (float)i - (float)(NUM_SAMPLES / 2);
        float ka = fabsf(k);
        float ks = (ka == 0.0f) ? 1.0f : ka;
        float warped;
        if (k == 0.0f)      warped = 0.0f;
        else if (k >= 0.0f) warped = powf(ks, beta_f) * stride;
        else                warped = -(powf(ks, beta_b) * stride);
        float env = (k >= 0.0f) ? strength / powf(1.0f + ka, alpha_f)
                                : strength / powf(1.0f + ka, alpha_b);
        float rw = env * (1.0f + tanhf(bias[i]));
        float c  = deformation + warped;
        hdr[i]        = warped;
        hdr[16 + i]   = rw;
        hdri[32 + i]  = (int)floorf(c);
        cmax = fmaxf(cmax, fabsf(c));
    }
    hdr[48]  = deformation;
    hdri[49] = (int)ceilf(cmax) + 4;
}

__global__ __launch_bounds__(256)
void build_map_kernel(const float* __restrict__ hdr, float* __restrict__ nw,
                      float* __restrict__ partials, int L) {
    __shared__ float s_w[NUM_SAMPLES];
    __shared__ float s_rw[NUM_SAMPLES];
    __shared__ int   s_F[NUM_SAMPLES];
    __shared__ float s_red[256];
    const int tid = threadIdx.x;
    if (tid < NUM_SAMPLES) {
        s_w[tid]  = hdr[tid];
        s_rw[tid] = hdr[16 + tid];
        s_F[tid]  = ((const int*)hdr)[32 + tid];
    }
    __syncthreads();
    const float d     = hdr[48];
    const int   Mloop = ((const int*)hdr)[49];
    const int   t     = blockIdx.x * blockDim.x + tid;
    const float Lm1   = (float)(L - 1);
    float wm = 0.0f, hit = 0.0f;
    if (t < L) {
        if (t == 0 || t == L - 1) {
            // boundary: clipping piles mass onto index 0 / L-1; brute force window
            int lo = (t == 0) ? 0 : max(0, L - 1 - Mloop);
            int hi = (t == 0) ? min(L - 1, Mloop) : (L - 1);
            for (int l = lo; l <= hi; ++l) {
                float centers = (float)l + d;               // reference fp order
                for (int k = 0; k < NUM_SAMPLES; ++k) {
                    float p  = centers + s_w[k];
                    p        = fminf(fmaxf(p, 0.0f), Lm1);
                    float pf = floorf(p);
                    int   fi = (int)pf;
                    int   ci = min(fi + 1, L - 1);
                    float fr = p - pf;
                    if (fi == t) { wm += (1.0f - fr) * s_rw[k]; hit += (1.0f - fr); }
                    if (ci == t) { wm += fr * s_rw[k];          hit += fr; }
                }
            }
        } else {
            // interior: exact gather inversion of the bilinear splat
            for (int k = 0; k < NUM_SAMPLES; ++k) {
                const int base = t - s_F[k];
                #pragma unroll
                for (int dl = -2; dl <= 1; ++dl) {
                    int l = base + dl;
                    if (l < 0 || l > L - 1) continue;
                    float p  = ((float)l + d) + s_w[k];     // reference fp order
                    p        = fminf(fmaxf(p, 0.0f), Lm1);
                    float pf = floorf(p);
                    int   fi = (int)pf;
                    int   ci = min(fi + 1, L - 1);
                    float fr = p - pf;
                    if (fi == t) { wm += (1.0f - fr) * s_rw[k]; hit += (1.0f - fr); }
                    if (ci == t) { wm += fr * s_rw[k];          hit += fr; }
                }
            }
        }
        float avg = wm / fmaxf(hit, 1e-6f);
        nw[t]      = avg;
        s_red[tid] = avg;
    } else {
        s_red[tid] = 0.0f;
    }
    __syncthreads();
    for (int off = 128; off > 0; off >>= 1) {
        if (tid < off) s_red[tid] += s_red[tid + off];
        __syncthreads();
    }
    if (tid == 0) partials[blockIdx.x] = s_red[0];
}

__global__ __launch_bounds__(256)
void reduce_kernel(const float* __restrict__ partials, int n, float* __restrict__ hdr) {
    __shared__ float s_red[256];
    float acc = 0.0f;
    for (int i = threadIdx.x; i < n; i += 256) acc += partials[i];  // fixed order
    s_red[threadIdx.x] = acc;
    __syncthreads();
    for (int off = 128; off > 0; off >>= 1) {
        if (threadIdx.x < off) s_red[threadIdx.x] += s_red[threadIdx.x + off];
        __syncthreads();
    }
    if (threadIdx.x == 0) hdr[51] = 1.0f / fmaxf(s_red[0], 1e-6f);
}

__global__ __launch_bounds__(256)
void scale_kernel(float* __restrict__ nw, const float* __restrict__ hdr, int L) {
    int t = blockIdx.x * blockDim.x + threadIdx.x;
    if (t < L) nw[t] = nw[t] * hdr[51];
}

// Tiled apply: block = (column chunk, row group). Stage the 4 KB nw chunk into
// LDS once with the CDNA5 async-copy path (ASYNCcnt), hoist to VGPRs, then
// stream ROWGRP rows of x/out with non-temporal b128 accesses.
__global__ __launch_bounds__(256)
void apply_tiled_kernel(const float* __restrict__ x, const float* __restrict__ nw,
                        float* __restrict__ out, int L) {
    __shared__ float s_nw[CHUNK];
    const int tid   = threadIdx.x;
    const int chunk = blockIdx.x;
    const int rowg  = blockIdx.y;
    const float* nwc = nw + (size_t)chunk * CHUNK;

    // each lane copies 16 bytes: from byte offset g_off of nwc into LDS at lds_off
    unsigned lds_off = (unsigned)(size_t)(&s_nw[0]) + (unsigned)tid * 16u;
    unsigned g_off   = (unsigned)tid * 16u;
    asm volatile("global_load_async_to_lds_b128 %0, %1, %2"
                 :: "v"(lds_off), "v"(g_off), "s"(nwc) : "memory");
    asm volatile("s_wait_asynccnt 0x0" ::: "memory");
    __syncthreads();

    const vfloat4* s_nw4 = (const vfloat4*)s_nw;
    const vfloat4  wv    = s_nw4[tid];           // reused across all rows

    const vfloat4* x4 = (const vfloat4*)x;
    vfloat4*       o4 = (vfloat4*)out;
    const size_t rowstride4 = (size_t)L / 4;
    size_t idx = (size_t)(rowg * ROWGRP) * rowstride4
               + (size_t)chunk * (CHUNK / 4) + (size_t)tid;
    #pragma unroll 4
    for (int r = 0; r < ROWGRP; ++r) {
        vfloat4 xv = __builtin_nontemporal_load(&x4[idx]);
        __builtin_nontemporal_store(xv * wv, &o4[idx]);
        idx += rowstride4;
    }
}

// fallback for shapes that don't tile (not taken for B=64, L=262144)
__global__ __launch_bounds__(256)
void apply_kernel(const float* __restrict__ x, const float* __restrict__ nw,
                  float* __restrict__ out, int L4mask, int total4) {
    int i = blockIdx.x * blockDim.x + threadIdx.x;
    if (i >= total4) return;
    const vfloat4* x4  = (const vfloat4*)x;
    const vfloat4* nw4 = (const vfloat4*)nw;
    vfloat4*       o4  = (vfloat4*)out;
    vfloat4 xv = __builtin_nontemporal_load(&x4[i]);
    vfloat4 wv = nw4[i & L4mask];
    __builtin_nontemporal_store(xv * wv, &o4[i]);
}

extern "C" void kernel_launch(void* const* d_in, const int* in_sizes, int n_in,
                              void* d_out, int out_size, void* d_ws, size_t ws_size,
                              hipStream_t stream) {
    (void)n_in; (void)ws_size;
    const float* x    = (const float*)d_in[0];
    const float* bd   = (const float*)d_in[1];
    const float* bs   = (const float*)d_in[2];
    const float* bbf  = (const float*)d_in[3];
    const float* bbb  = (const float*)d_in[4];
    const float* bstr = (const float*)d_in[5];
    const float* baf  = (const float*)d_in[6];
    const float* bab  = (const float*)d_in[7];
    const float* bias = (const float*)d_in[8];

    const int L = in_sizes[0] / BATCH;          // 262144
    const int B = BATCH;
    float* hdr      = (float*)d_ws;
    float* nw       = hdr + 64;                 // 256B-aligned
    const int nBlk  = (L + 255) / 256;          // 1024
    float* partials = nw + L;

    prep_kernel<<<1, 32, 0, stream>>>(bd, bs, bbf, bbb, bstr, baf, bab, bias, hdr);
    build_map_kernel<<<nBlk, 256, 0, stream>>>(hdr, nw, partials, L);
    reduce_kernel<<<1, 256, 0, stream>>>(partials, nBlk, hdr);
    scale_kernel<<<nBlk, 256, 0, stream>>>(nw, hdr, L);

    if ((L % CHUNK) == 0 && (B % ROWGRP) == 0) {
        dim3 grid(L / CHUNK, B / ROWGRP);       // (256, 4) = 1024 blocks
        apply_tiled_kernel<<<grid, 256, 0, stream>>>(x, nw, (float*)d_out, L);
    } else {
        const int total4 = out_size / 4;
        apply_kernel<<<(total4 + 255) / 256, 256, 0, stream>>>(
            x, nw, (float*)d_out, (L / 4) - 1, total4);
    }
}